// Attention_73194832658757
// MI455X (gfx1250) — compile-verified
//
#include <hip/hip_runtime.h>
#include <hip/hip_bf16.h>
#include <math.h>

// Problem constants (reference: B=4, S=4096, D=32)
constexpr int B = 4;
constexpr int S = 4096;
constexpr int D = 32;
constexpr int JTILES = S / 16;   // 256
constexpr int ITILES = S / 16;   // 256
constexpr int NCH = 8;           // i-chunks for column-stats pass
constexpr int NJC = 8;           // j-chunks for attention pass
// q is pre-scaled by (1/sqrt(D)) * log2(e): softmax runs in exp2 domain
// (softmax is invariant to the base change, saves a v_mul per element).
constexpr float QSCALE = 0.17677669529663687f * 1.4426950408889634f;

typedef float v2f __attribute__((ext_vector_type(2)));
typedef float v8f __attribute__((ext_vector_type(8)));

// CDNA5 fp32 WMMA: D(16x16,f32) = A(16x4,f32) * B(4x16,f32) + C
__device__ __forceinline__ v8f wmma_f32_16x16x4(v2f a, v2f b, v8f c) {
  return __builtin_amdgcn_wmma_f32_16x16x4_f32(false, a, false, b,
                                               (short)0, c, false, false);
}

__device__ __forceinline__ float wave_sum32(float x) {
  #pragma unroll
  for (int off = 16; off > 0; off >>= 1) x += __shfl_xor(x, off, 32);
  return x;
}

// ---------------------------------------------------------------------------
// Kernel 1: fused LayerNorm + 32x32 projection. One wave32 per row (lane==d).
// postScale folds the score scale + log2e into the q projection.
// ---------------------------------------------------------------------------
__global__ void ln_proj_kernel(const float* __restrict__ X,
                               const float* __restrict__ g,
                               const float* __restrict__ bln,
                               const float* __restrict__ W,
                               const float* __restrict__ bias,
                               float* __restrict__ Y,
                               float postScale) {
  const int row  = blockIdx.x * 8 + (threadIdx.x >> 5);
  const int lane = threadIdx.x & 31;
  float x  = X[(size_t)row * D + lane];
  float m  = wave_sum32(x) * (1.0f / 32.0f);
  float dv = x - m;
  float var = wave_sum32(dv * dv) * (1.0f / 32.0f);
  float xn = dv * rsqrtf(var + 1e-5f) * g[lane] + bln[lane];
  float y = bias[lane];
  #pragma unroll
  for (int k = 0; k < 32; ++k)
    y += __shfl(xn, k, 32) * W[k * D + lane];
  Y[(size_t)row * D + lane] = y * postScale;
}

// Per-lane online (max, sum-exp2) update for one C-fragment column slice.
// Lane owns column j, rows i0 + v + 8h. No cross-lane traffic here.
__device__ __forceinline__ void online_update(const v8f& c,
                                              const unsigned char* __restrict__ mcol,
                                              int i0, int h,
                                              float& m_run, float& l_run) {
  float sv[8];
  float tm = -__builtin_inff();
  #pragma unroll
  for (int v = 0; v < 8; ++v) {
    const unsigned char mk = mcol[(size_t)(i0 + v + 8 * h) * S];
    sv[v] = mk ? -__builtin_inff() : c[v];
    tm = fmaxf(tm, sv[v]);
  }
  const float mnew = fmaxf(m_run, tm);
  const float sm   = (mnew > -1e37f) ? mnew : 0.0f;
  float ts = 0.0f;
  #pragma unroll
  for (int v = 0; v < 8; ++v) ts += exp2f(sv[v] - sm);
  const float corr = (m_run > -1e37f) ? exp2f(m_run - sm) : 0.0f;
  l_run = l_run * corr + ts;
  m_run = mnew;
}

// Merge (m,l) pairs across the two lane-halves (rows 0-7 vs 8-15).
__device__ __forceinline__ void merge_halves(float& m, float& l) {
  const float mo = __shfl_xor(m, 16, 32);
  const float lo = __shfl_xor(l, 16, 32);
  const float mm = fmaxf(m, mo);
  const float sm = (mm > -1e37f) ? mm : 0.0f;
  const float c0 = (m  > -1e37f) ? exp2f(m  - sm) : 0.0f;
  const float c1 = (mo > -1e37f) ? exp2f(mo - sm) : 0.0f;
  l = l * c0 + lo * c1;
  m = mm;
}

// ---------------------------------------------------------------------------
// Kernel 2: per-column (softmax over the query axis) online max / sum-exp2.
// One wave per PAIR of 16-column j-tiles (q A-fragments amortized over 16
// WMMA per i-step); blockIdx.y selects an i-chunk (partial stats).
// ---------------------------------------------------------------------------
__global__ void colstats_kernel(const float* __restrict__ qp,
                                const float* __restrict__ kp,
                                const unsigned char* __restrict__ mask,
                                float* __restrict__ cmax_p,
                                float* __restrict__ csum_p) {
  const int wave = threadIdx.x >> 5;
  const int lane = threadIdx.x & 31;
  const int jp   = blockIdx.x * 4 + wave;        // [0, B*JTILES/2)
  const int b    = jp / (JTILES / 2);
  const int j0   = (jp % (JTILES / 2)) * 32;     // two adjacent 16-col tiles
  const int chunk = blockIdx.y;
  const int iBeg = chunk * (S / NCH);
  const int iEnd = iBeg + (S / NCH);
  const int n16  = lane & 15;
  const int h    = lane >> 4;

  // B fragments for both j-tiles, resident for the whole i loop.
  v2f kb0[8], kb1[8];
  const float* krow0 = kp + ((size_t)b * S + j0 + n16) * D;
  #pragma unroll
  for (int kk = 0; kk < 8; ++kk) {
    kb0[kk] = *(const v2f*)(krow0 + 4 * kk + 2 * h);
    kb1[kk] = *(const v2f*)(krow0 + 16 * D + 4 * kk + 2 * h);
  }

  float m0 = -__builtin_inff(), l0 = 0.0f;
  float m1 = -__builtin_inff(), l1 = 0.0f;
  const size_t mbase = (size_t)b * S * S;
  const unsigned char* mcol0 = mask + mbase + j0 + n16;

  for (int i0 = iBeg; i0 < iEnd; i0 += 16) {
    // Prefetch mask 4 tiles ahead: each lane pulls one row's cacheline.
    __builtin_prefetch(mask + mbase + (size_t)(i0 + 64 + n16 + 16 * h) * S + j0, 0, 1);
    const float* qrow = qp + ((size_t)b * S + i0 + n16) * D;
    v8f c0 = {}, c1 = {};
    #pragma unroll
    for (int kk = 0; kk < 8; ++kk) {
      v2f a = *(const v2f*)(qrow + 4 * kk + 2 * h);
      c0 = wmma_f32_16x16x4(a, kb0[kk], c0);
      c1 = wmma_f32_16x16x4(a, kb1[kk], c1);
    }
    online_update(c0, mcol0,      i0, h, m0, l0);
    online_update(c1, mcol0 + 16, i0, h, m1, l1);
  }
  merge_halves(m0, l0);
  merge_halves(m1, l1);
  if (lane < 16) {
    const size_t base = ((size_t)chunk * B + b) * S;
    cmax_p[base + j0 + lane]      = m0;
    csum_p[base + j0 + lane]      = l0;
    cmax_p[base + j0 + 16 + lane] = m1;
    csum_p[base + j0 + 16 + lane] = l1;
  }
}

// ---------------------------------------------------------------------------
// Kernel 2b: deterministic combine of i-chunk partial (max, sumexp2).
// ---------------------------------------------------------------------------
__global__ void colcombine_kernel(const float* __restrict__ cmax_p,
                                  const float* __restrict__ csum_p,
                                  float* __restrict__ cmax,
                                  float* __restrict__ csum) {
  const int idx = blockIdx.x * blockDim.x + threadIdx.x;  // [0, B*S)
  if (idx >= B * S) return;
  float m = -__builtin_inff();
  #pragma unroll
  for (int c = 0; c < NCH; ++c) m = fmaxf(m, cmax_p[(size_t)c * B * S + idx]);
  const float sm = (m > -1e37f) ? m : 0.0f;
  float l = 0.0f;
  #pragma unroll
  for (int c = 0; c < NCH; ++c) {
    const float mc = cmax_p[(size_t)c * B * S + idx];
    const float corr = (mc > -1e37f) ? exp2f(mc - sm) : 0.0f;
    l += csum_p[(size_t)c * B * S + idx] * corr;
  }
  cmax[idx] = m;
  csum[idx] = l;
}

// ---------------------------------------------------------------------------
// Kernel 3: attn = probs @ v. One wave per PAIR of 16-row i-tiles (k and v
// fragments amortized over 32 WMMA per j-step -> WMMA-bound); blockIdx.y =
// j-chunk. Score tiles recomputed with WMMA, p transposed C->A layout through
// LDS (pitch 18 keeps float2 LDS loads 8B-aligned), p@v accumulated with WMMA.
// ---------------------------------------------------------------------------
__global__ void attn_kernel(const float* __restrict__ qp,
                            const float* __restrict__ kp,
                            const float* __restrict__ vp,
                            const unsigned char* __restrict__ mask,
                            const float* __restrict__ cmax,
                            const float* __restrict__ csum,
                            float* __restrict__ attn_p) {
  __shared__ float pls[4][2 * 16 * 18];
  const int wave = threadIdx.x >> 5;
  const int lane = threadIdx.x & 31;
  const int ip   = blockIdx.x * 4 + wave;        // [0, B*ITILES/2)
  const int b    = ip / (ITILES / 2);
  const int i0   = (ip % (ITILES / 2)) * 32;     // rows i0..i0+31 (two tiles)
  const int jc   = blockIdx.y;
  const int jBeg = jc * (S / NJC);
  const int jEnd = jBeg + (S / NJC);
  const int n16  = lane & 15;
  const int h    = lane >> 4;
  float* pw = pls[wave];

  // q A-fragments for both i-tiles are loop-invariant.
  v2f qa0[8], qa1[8];
  const float* qrow = qp + ((size_t)b * S + i0 + n16) * D;
  #pragma unroll
  for (int kk = 0; kk < 8; ++kk) {
    qa0[kk] = *(const v2f*)(qrow + 4 * kk + 2 * h);
    qa1[kk] = *(const v2f*)(qrow + 16 * D + 4 * kk + 2 * h);
  }

  v8f acc00 = {}, acc01 = {}, acc10 = {}, acc11 = {};  // [itile][dtile]
  const size_t mbase = (size_t)b * S * S;

  for (int j0 = jBeg; j0 < jEnd; j0 += 16) {
    // Prefetch mask 4 j-steps ahead: each lane pulls one of the 32 rows.
    __builtin_prefetch(mask + mbase + (size_t)(i0 + n16 + 16 * h) * S + j0 + 64, 0, 1);
    // --- score tiles s = q . k^T (already in exp2 domain) ---
    const float* krow = kp + ((size_t)b * S + j0 + n16) * D;
    v8f c0 = {}, c1 = {};
    #pragma unroll
    for (int kk = 0; kk < 8; ++kk) {
      v2f bf = *(const v2f*)(krow + 4 * kk + 2 * h);
      c0 = wmma_f32_16x16x4(qa0[kk], bf, c0);
      c1 = wmma_f32_16x16x4(qa1[kk], bf, c1);
    }
    // --- p = exp2(s - colmax[j]) / colsum[j], 0 where masked ---
    const int j     = j0 + n16;
    const float cm  = cmax[(size_t)b * S + j];
    const float cl  = csum[(size_t)b * S + j];
    const float rcl = (cl > 0.0f) ? 1.0f / cl : 0.0f;
    const float scm = (cm > -1e37f) ? cm : 0.0f;
    #pragma unroll
    for (int v = 0; v < 8; ++v) {
      const int iv0 = i0 + v + 8 * h;
      const unsigned char mk0 = mask[mbase + (size_t)iv0 * S + j];
      const unsigned char mk1 = mask[mbase + (size_t)(iv0 + 16) * S + j];
      const float p0 = mk0 ? 0.0f : exp2f(c0[v] - scm) * rcl;
      const float p1 = mk1 ? 0.0f : exp2f(c1[v] - scm) * rcl;
      pw[(v + 8 * h) * 18 + n16]       = p0;   // tile0: [row i][col j]
      pw[288 + (v + 8 * h) * 18 + n16] = p1;   // tile1
    }
    // DS ops of a wave are in-order; barrier keeps compiler from reordering
    // and drains DScnt before re-reading the tiles as A-fragments.
    asm volatile("s_wait_dscnt 0" ::: "memory");
    // --- acc += p @ v  (two i-tiles x two 16-wide d-tiles) ---
    #pragma unroll
    for (int kk = 0; kk < 4; ++kk) {
      v2f pa0 = *(const v2f*)(pw + n16 * 18 + 4 * kk + 2 * h);
      v2f pa1 = *(const v2f*)(pw + 288 + n16 * 18 + 4 * kk + 2 * h);
      const float* vrow = vp + ((size_t)b * S + j0 + 4 * kk + 2 * h) * D;
      v2f vb0, vb1;
      vb0.x = vrow[n16];        vb0.y = vrow[D + n16];
      vb1.x = vrow[16 + n16];   vb1.y = vrow[D + 16 + n16];
      acc00 = wmma_f32_16x16x4(pa0, vb0, acc00);
      acc01 = wmma_f32_16x16x4(pa0, vb1, acc01);
      acc10 = wmma_f32_16x16x4(pa1, vb0, acc10);
      acc11 = wmma_f32_16x16x4(pa1, vb1, acc11);
    }
    asm volatile("" ::: "memory");  // keep next tile's DS stores behind loads
  }

  // Store both partial attention tiles (C layout).
  float* outp = attn_p + ((size_t)jc * B * S + (size_t)b * S + i0) * D;
  #pragma unroll
  for (int v = 0; v < 8; ++v) {
    outp[(v + 8 * h) * D + n16]             = acc00[v];
    outp[(v + 8 * h) * D + 16 + n16]        = acc01[v];
    outp[(16 + v + 8 * h) * D + n16]        = acc10[v];
    outp[(16 + v + 8 * h) * D + 16 + n16]   = acc11[v];
  }
}

// ---------------------------------------------------------------------------
// Kernel 4: sum j-chunk partials, out = attn @ Wo + bo + residual(Q).
// ---------------------------------------------------------------------------
__global__ void final_kernel(const float* __restrict__ attn_p,
                             const float* __restrict__ Wo,
                             const float* __restrict__ bo,
                             const float* __restrict__ Q,
                             float* __restrict__ out) {
  const int row  = blockIdx.x * 8 + (threadIdx.x >> 5);   // [0, B*S)
  const int lane = threadIdx.x & 31;
  float a = 0.0f;
  #pragma unroll
  for (int c = 0; c < NJC; ++c)
    a += attn_p[((size_t)c * B * S + row) * D + lane];
  float y = bo[lane];
  #pragma unroll
  for (int k = 0; k < 32; ++k)
    y += __shfl(a, k, 32) * Wo[k * D + lane];
  out[(size_t)row * D + lane] = Q[(size_t)row * D + lane] + y;
}

// ---------------------------------------------------------------------------
extern "C" void kernel_launch(void* const* d_in, const int* in_sizes, int n_in,
                              void* d_out, int out_size, void* d_ws, size_t ws_size,
                              hipStream_t stream) {
  (void)in_sizes; (void)n_in; (void)out_size; (void)ws_size;
  const float* Q    = (const float*)d_in[0];
  const float* K    = (const float*)d_in[1];
  const float* V    = (const float*)d_in[2];
  const unsigned char* mask = (const unsigned char*)d_in[3];  // jnp bool = 1B
  const float* ln_g = (const float*)d_in[4];
  const float* ln_b = (const float*)d_in[5];
  const float* Wq   = (const float*)d_in[6];
  const float* bq   = (const float*)d_in[7];
  const float* Wk   = (const float*)d_in[8];
  const float* bk   = (const float*)d_in[9];
  const float* Wv   = (const float*)d_in[10];
  const float* bv   = (const float*)d_in[11];
  const float* Wo   = (const float*)d_in[12];
  const float* bo   = (const float*)d_in[13];
  float* out = (float*)d_out;

  // Workspace carve-up (floats): qkv (3x2MB) + partial/final col stats
  // + j-chunk partial attn (8x2MB) -> ~23 MB. Deterministic (no FP atomics).
  constexpr size_t QKV  = (size_t)B * S * D;
  constexpr size_t COLS = (size_t)B * S;
  float* ws     = (float*)d_ws;
  float* qp     = ws;
  float* kp     = qp + QKV;
  float* vp     = kp + QKV;
  float* cmax_p = vp + QKV;
  float* csum_p = cmax_p + NCH * COLS;
  float* cmax   = csum_p + NCH * COLS;
  float* csum   = cmax + COLS;
  float* attn_p = csum + COLS;

  const int rows = B * S;                 // 16384
  // Pass A: LN + projections (wave per row); q pre-scaled into exp2 domain
  ln_proj_kernel<<<rows / 8, 256, 0, stream>>>(Q, ln_g, ln_b, Wq, bq, qp, QSCALE);
  ln_proj_kernel<<<rows / 8, 256, 0, stream>>>(K, ln_g, ln_b, Wk, bk, kp, 1.0f);
  ln_proj_kernel<<<rows / 8, 256, 0, stream>>>(V, ln_g, ln_b, Wv, bv, vp, 1.0f);
  // Pass B: column softmax stats (softmax over query axis), i-chunked
  colstats_kernel<<<dim3(B * JTILES / 8, NCH), 128, 0, stream>>>(
      qp, kp, mask, cmax_p, csum_p);
  colcombine_kernel<<<(B * S + 255) / 256, 256, 0, stream>>>(
      cmax_p, csum_p, cmax, csum);
  // Pass C: attn partials per j-chunk (WMMA score recompute + WMMA p@v)
  attn_kernel<<<dim3(B * ITILES / 8, NJC), 128, 0, stream>>>(
      qp, kp, vp, mask, cmax, csum, attn_p);
  // Pass D: reduce partials + output projection + residual
  final_kernel<<<rows / 8, 256, 0, stream>>>(attn_p, Wo, bo, Q, out);
}